// YOLOV5TorchObjectDetector_16612933501393
// MI455X (gfx1250) — compile-verified
//
#include <hip/hip_runtime.h>
#include <hip/hip_bf16.h>
#include <stdint.h>

// ---------------------------------------------------------------------------
// YOLOv5 NMS post-processing for MI455X (gfx1250).
//  k_preprocess : bandwidth-bound streaming pass (137 MB read @ 23.3 TB/s
//                 => ~11 us floor; b128 load clauses + dual-issue VALU).
//  k_nms        : 1 workgroup / image (16 WGPs); 25200 scores live in 100 KB
//                 of CDNA5 LDS, staged by one Tensor Data Mover descriptor;
//                 300 serial argmax+IoU steps with wave32 shuffle reductions;
//                 boxes stay L2-resident (192 MB L2), read as float4/b128.
//  k_gather     : emits [16,300,87] rows, zeroing invalid detections.
// No matmul structure exists in NMS (max/argmax/min-max), so WMMA does not
// apply; the CDNA5-specific paths used are TDM, s_wait_tensorcnt, 320KB LDS,
// and global_prefetch_b8.
// ---------------------------------------------------------------------------

#define BATCH     16
#define NPRED     25200
#define NCLS      80
#define CCH       (NCLS + 5)   // 85
#define MAX_DET   300
#define OUT_C     87
#define CONF_THR  0.25f
#define IOU_THR   0.45f
#define NMS_BLOCK 1024

typedef unsigned int v4u __attribute__((ext_vector_type(4)));
typedef int          v8i __attribute__((ext_vector_type(8)));
typedef int          v4i __attribute__((ext_vector_type(4)));

#ifndef __has_builtin
#define __has_builtin(x) 0
#endif

#if __has_builtin(__builtin_amdgcn_tensor_load_to_lds)
#define USE_TDM 1
#endif

// ---------------------------------------------------------------------------
// Kernel 1: per-prediction preprocessing (class max, xywh->xyxy, score).
// ---------------------------------------------------------------------------
__global__ void k_preprocess(const float* __restrict__ pred,
                             float* __restrict__ boxes,    // [B*N,4]
                             float* __restrict__ scores)   // [B*N]
{
    int i = blockIdx.x * blockDim.x + threadIdx.x;
    if (i >= BATCH * NPRED) return;
    const float* p = pred + (size_t)i * CCH;

    // CDNA5 speculative prefetch a few rows ahead -> global_prefetch_b8
    __builtin_prefetch(p + (size_t)CCH * 64, 0, 1);

    float x = p[0], y = p[1], w = p[2], h = p[3], obj = p[4];

    float best = -__builtin_inff();
    #pragma unroll 16
    for (int c = 0; c < NCLS; ++c) {
        float v = p[5 + c] * obj;
        if (v > best) { best = v; }     // cls argmax recomputed in k_gather
    }

    float x1 = x - 0.5f * w, y1 = y - 0.5f * h;
    float x2 = x + 0.5f * w, y2 = y + 0.5f * h;
    float4 bb = make_float4(x1, y1, x2, y2);
    reinterpret_cast<float4*>(boxes)[i] = bb;       // global_store_b128

    bool valid = (obj > CONF_THR) && (best > CONF_THR);
    scores[i] = valid ? best : -__builtin_inff();
}

// ---------------------------------------------------------------------------
// Kernel 2: sequential greedy NMS, one workgroup per image.
// ---------------------------------------------------------------------------
__global__ __launch_bounds__(NMS_BLOCK)
void k_nms(const float* __restrict__ boxes,     // [B*N,4]
           const float* __restrict__ scores_g,  // [B*N]
           int* __restrict__ sel,               // [B,MAX_DET]
           int* __restrict__ okf)               // [B,MAX_DET]
{
    __shared__ float s_scores[NPRED];           // 100,800 B of 320 KB LDS
    __shared__ float s_rv[32];
    __shared__ int   s_ri[32];
    __shared__ float s_box[5];                  // x1,y1,x2,y2,area
    __shared__ int   s_ok;

    const int b   = blockIdx.x;
    const int tid = threadIdx.x;
    const float4* bx4 = reinterpret_cast<const float4*>(boxes) + (size_t)b * NPRED;
    const float*  sg  = scores_g + (size_t)b * NPRED;

    // ---- stage scores into LDS via Tensor Data Mover (1 DMA descriptor) ----
#ifdef USE_TDM
    if (tid < 32) {  // wave 0 issues the TDM op (EXEC ignored by TDM)
        uint64_t ga  = (uint64_t)(uintptr_t)sg;
        uint32_t lds = (uint32_t)(uintptr_t)(void*)&s_scores[0]; // low 32b = LDS offset

        v4u g0;
        g0.x = 1u;                                   // count=1 valid descriptor
        g0.y = lds;                                  // lds_addr
        g0.z = (uint32_t)ga;                         // global_addr[31:0]
        g0.w = (uint32_t)((ga >> 32) & 0x01FFFFFFu)  // global_addr[56:32]
             | 0x80000000u;                          // type=2 ("image")
        v8i g1;
        g1[0] = 0x00020000;                          // data_size=2 (4 bytes)
        g1[1] = (int)((NPRED & 0xFFFF) << 16);       // tensor_dim0[15:0]
        g1[2] = (int)((NPRED >> 16) | (1u << 16));   // tensor_dim0[31:16] | tensor_dim1=1
        g1[3] = (int)((NPRED & 0xFFFF) << 16);       // tile_dim0 = 25200
        g1[4] = 1;                                   // tile_dim1=1, tile_dim2=0
        g1[5] = NPRED;                               // tensor_dim0_stride[31:0]
        g1[6] = 0;
        g1[7] = 0;
        v4i gz = {0, 0, 0, 0};
#if defined(__clang_major__) && (__clang_major__ >= 23)
        v8i gz8 = {0, 0, 0, 0, 0, 0, 0, 0};
        __builtin_amdgcn_tensor_load_to_lds(g0, g1, gz, gz, gz8, 0);
#else
        __builtin_amdgcn_tensor_load_to_lds(g0, g1, gz, gz, 0);
#endif
        __builtin_amdgcn_s_wait_tensorcnt(0);        // -> s_wait_tensorcnt
    }
#else
    for (int j = tid; j < NPRED; j += NMS_BLOCK) s_scores[j] = sg[j];
#endif
    __syncthreads();

    const int lane = tid & 31;
    const int wid  = tid >> 5;

    for (int it = 0; it < MAX_DET; ++it) {
        // ---- block-wide argmax (first-occurrence ties, like jnp.argmax) ----
        float bv = s_scores[tid];
        int   bi = tid;
        for (int j = tid + NMS_BLOCK; j < NPRED; j += NMS_BLOCK) {
            float v = s_scores[j];
            if (v > bv) { bv = v; bi = j; }
        }
        #pragma unroll
        for (int off = 16; off > 0; off >>= 1) {
            float ov = __shfl_down(bv, off, 32);
            int   oi = __shfl_down(bi, off, 32);
            if (ov > bv || (ov == bv && oi < bi)) { bv = ov; bi = oi; }
        }
        if (lane == 0) { s_rv[wid] = bv; s_ri[wid] = bi; }
        __syncthreads();
        if (wid == 0) {
            bv = s_rv[lane];
            bi = s_ri[lane];
            #pragma unroll
            for (int off = 16; off > 0; off >>= 1) {
                float ov = __shfl_down(bv, off, 32);
                int   oi = __shfl_down(bi, off, 32);
                if (ov > bv || (ov == bv && oi < bi)) { bv = ov; bi = oi; }
            }
            if (lane == 0) {
                int ok = (bv > -__builtin_inff()) ? 1 : 0;
                s_ok = ok;
                sel[b * MAX_DET + it] = bi;
                okf[b * MAX_DET + it] = ok;
                float4 B = bx4[bi];                       // global_load_b128
                s_box[0] = B.x; s_box[1] = B.y;
                s_box[2] = B.z; s_box[3] = B.w;
                s_box[4] = (B.z - B.x) * (B.w - B.y);
            }
        }
        __syncthreads();

        // ---- IoU suppression (boxes L2-resident, b128 per candidate) ----
        if (s_ok) {
            float X1 = s_box[0], Y1 = s_box[1], X2 = s_box[2], Y2 = s_box[3];
            float A  = s_box[4];
            for (int j = tid; j < NPRED; j += NMS_BLOCK) {
                float4 J = bx4[j];                        // global_load_b128
                float lx = fmaxf(X1, J.x), ly = fmaxf(Y1, J.y);
                float rx = fminf(X2, J.z), ry = fminf(Y2, J.w);
                float iw = fmaxf(rx - lx, 0.0f), ih = fmaxf(ry - ly, 0.0f);
                float inter = iw * ih;
                float aj  = (J.z - J.x) * (J.w - J.y);
                float iou = inter / (A + aj - inter + 1e-9f);
                if (iou > IOU_THR) s_scores[j] = -__builtin_inff();
            }
        }
        __syncthreads();
    }
}

// ---------------------------------------------------------------------------
// Kernel 3: gather selected rows -> [B, MAX_DET, 87]; zero invalid rows.
// Row layout: [x1,y1,x2,y2, conf, cls, obj, logits(80)].
// ---------------------------------------------------------------------------
__global__ void k_gather(const float* __restrict__ pred,
                         const float* __restrict__ logits,
                         const float* __restrict__ boxes,
                         const int* __restrict__ sel,
                         const int* __restrict__ okf,
                         float* __restrict__ out)
{
    int t = blockIdx.x * blockDim.x + threadIdx.x;
    if (t >= BATCH * MAX_DET * OUT_C) return;
    int c = t % OUT_C;
    int d = (t / OUT_C) % MAX_DET;
    int b = t / (OUT_C * MAX_DET);

    int   idx = sel[b * MAX_DET + d];
    float ok  = (float)okf[b * MAX_DET + d];
    size_t row = (size_t)b * NPRED + (size_t)idx;

    float v;
    if (c < 4) {
        v = boxes[row * 4 + c];
    } else if (c == 6) {
        v = pred[row * CCH + 4];                       // objectness
    } else if (c == 4 || c == 5) {
        const float* p = pred + row * CCH;
        float obj = p[4];
        float best = -__builtin_inff(); int bc = 0;
        #pragma unroll 16
        for (int k = 0; k < NCLS; ++k) {
            float vv = p[5 + k] * obj;
            if (vv > best) { best = vv; bc = k; }
        }
        v = (c == 4) ? best : (float)bc;
    } else {
        v = logits[row * NCLS + (c - 7)];
    }
    out[t] = v * ok;
}

// ---------------------------------------------------------------------------
extern "C" void kernel_launch(void* const* d_in, const int* in_sizes, int n_in,
                              void* d_out, int out_size, void* d_ws, size_t ws_size,
                              hipStream_t stream)
{
    const float* pred   = (const float*)d_in[0];   // [16,25200,85]
    const float* logits = (const float*)d_in[1];   // [16,25200,80]
    float* out = (float*)d_out;                    // [16,300,87]

    char* ws = (char*)d_ws;
    const size_t n = (size_t)BATCH * NPRED;
    float* boxes  = (float*)ws;  ws += n * 4 * sizeof(float);   // 6.45 MB
    float* scores = (float*)ws;  ws += n * sizeof(float);       // 1.61 MB
    int*   sel    = (int*)ws;    ws += (size_t)BATCH * MAX_DET * sizeof(int);
    int*   okf    = (int*)ws;

    {
        int threads = 256;
        int blocks  = (int)((n + threads - 1) / threads);
        k_preprocess<<<blocks, threads, 0, stream>>>(pred, boxes, scores);
    }
    k_nms<<<BATCH, NMS_BLOCK, 0, stream>>>(boxes, scores, sel, okf);
    {
        int total   = BATCH * MAX_DET * OUT_C;
        int threads = 256;
        k_gather<<<(total + threads - 1) / threads, threads, 0, stream>>>(
            pred, logits, boxes, sel, okf, out);
    }
}